// SwitchLinear_5033701671494
// MI455X (gfx1250) — compile-verified
//
#include <hip/hip_runtime.h>
#include <hip/hip_bf16.h>

typedef __attribute__((ext_vector_type(2))) float v2f;
typedef __attribute__((ext_vector_type(4))) float v4f;
typedef __attribute__((ext_vector_type(8))) float v8f;

constexpr int BTOK = 4096;
constexpr int IN   = 256;
constexpr int OUT  = 256;
constexpr int E    = 16;
constexpr int BDIM = 512;           // 16 waves (wave32) -> fills one WGP
constexpr int CG   = 64;            // output columns per block (4 WMMA N-tiles)
constexpr int LDS_STRIDE = IN + 4;  // 260 floats: break 64-bank periodicity
constexpr int LDS_BYTES  = CG * LDS_STRIDE * (int)sizeof(float); // ~65 KB

// ---------------------------------------------------------------------------
// Kernel 1: bucket tokens by expert.  counts[e] and buckets[e][i] live in d_ws.
// Bucket order is non-deterministic but each token's output row is computed
// independently, so final output values are deterministic.
// ---------------------------------------------------------------------------
__global__ void bin_tokens_kernel(const int* __restrict__ route,
                                  int* __restrict__ counts,
                                  int* __restrict__ buckets) {
    int b = blockIdx.x * blockDim.x + threadIdx.x;
    if (b < BTOK) {
        int e = route[b];
        int p = atomicAdd(&counts[e], 1);
        buckets[e * BTOK + p] = b;
    }
}

// ---------------------------------------------------------------------------
// Kernel 2: fused grouped GEMM.
//   grid = (E, OUT/64); block = 512 threads = 16 waves.
//   Each block: one expert e, a 64-wide output-column group.
//   LDS holds Wc = W[e] + W_fact for those 64 output rows (64 x 256 f32),
//   loaded from HBM exactly once per block.
//   Each wave walks this expert's 16-token tiles; per K-step:
//     1x global_load_b64 (A) + 2x ds_load_2addr_b64 (4 B pairs) + 4x WMMA
//   -> 4x A-fragment reuse, full f32 precision via V_WMMA_F32_16X16X4_F32.
//   Tail rows clamp to a valid token (unconditional loads, unstored rows).
// ---------------------------------------------------------------------------
__global__ __launch_bounds__(BDIM) void moe_wmma_kernel(
    const float* __restrict__ input,   // [BTOK, IN]
    const float* __restrict__ weight,  // [E, OUT*IN]
    const float* __restrict__ wfact,   // [OUT*IN]
    const float* __restrict__ bias,    // [E, OUT]
    const float* __restrict__ bfact,   // [OUT]
    const int*   __restrict__ counts,  // [E]
    const int*   __restrict__ buckets, // [E, BTOK]
    float*       __restrict__ out)     // [BTOK, OUT]
{
    extern __shared__ float lds[];     // [CG][LDS_STRIDE]

    const int e   = blockIdx.x;
    const int ob  = blockIdx.y * CG;   // output-column base
    const int tid = threadIdx.x;

    // --- build combined weight tile in LDS: lds[m][n] = W[e][ob+m][n] + Wf[ob+m][n]
    const float* we = weight + (size_t)e * (OUT * IN) + (size_t)ob * IN;
    const float* wf = wfact + (size_t)ob * IN;
    for (int j = tid; j < CG * (IN / 4); j += BDIM) {   // 4096 float4s
        int m = j >> 6;            // 64 float4 per row
        int n = (j & 63) << 2;
        v4f a = *(const v4f*)(we + m * IN + n);
        v4f b = *(const v4f*)(wf + m * IN + n);
        *(v4f*)(&lds[m * LDS_STRIDE + n]) = a + b;
    }
    __syncthreads();

    const int wave = tid >> 5;
    const int lane = tid & 31;
    const int half = lane >> 4;        // 0: lanes 0-15, 1: lanes 16-31
    const int l16  = lane & 15;

    const int  cnt    = counts[e];
    const int  ntiles = (cnt + 15) >> 4;
    const int* bkt    = buckets + e * BTOK;

    // Per N-tile (ct): lane owns column N = l16 -> global col ob + ct*16 + l16.
    float        bsum[4];
    const float* brow[4];
    #pragma unroll
    for (int ct = 0; ct < 4; ++ct) {
        int col  = ob + ct * 16 + l16;
        bsum[ct] = bias[e * OUT + col] + bfact[col];
        // B operand: row (ct*16 + l16) of Wc; lanes>=16 carry K+2,K+3.
        brow[ct] = &lds[(ct * 16 + l16) * LDS_STRIDE + (half << 1)];
    }

    for (int tile = wave; tile < ntiles; tile += BDIM / 32) {
        const int base = tile * 16;
        // A-layout: lane supplies row M = l16; clamp tail rows to last valid.
        int idx  = base + l16;
        idx      = idx < cnt ? idx : cnt - 1;
        int tokA = bkt[idx];
        const float* arow = input + (size_t)tokA * IN + (half << 1);

        v8f acc[4] = {{}, {}, {}, {}};
        #pragma unroll 8
        for (int k = 0; k < IN; k += 4) {
            v2f a = *(const v2f*)(arow + k);
            #pragma unroll
            for (int ct = 0; ct < 4; ++ct) {
                v2f b = *(const v2f*)(brow[ct] + k);
                acc[ct] = __builtin_amdgcn_wmma_f32_16x16x4_f32(
                    /*neg_a=*/false, a, /*neg_b=*/false, b,
                    /*c_mod=*/(short)0, acc[ct], /*reuse_a=*/false, /*reuse_b=*/false);
            }
        }

        // D-layout: VGPR g holds row M = g + 8*half, col N = l16 (per ct tile).
        // Row M's token id lives in lane M's tokA register -> shuffle, no reload.
        #pragma unroll
        for (int g = 0; g < 8; ++g) {
            int M = g + (half << 3);
            int r = base + M;
            int t = __shfl(tokA, M);
            if (r < cnt) {
                float* orow = out + (size_t)t * OUT + ob + l16;
                #pragma unroll
                for (int ct = 0; ct < 4; ++ct) {
                    orow[ct * 16] = acc[ct][g] + bsum[ct];
                }
            }
        }
    }
}

// ---------------------------------------------------------------------------
extern "C" void kernel_launch(void* const* d_in, const int* in_sizes, int n_in,
                              void* d_out, int out_size, void* d_ws, size_t ws_size,
                              hipStream_t stream) {
    const float* input  = (const float*)d_in[0];
    const int*   route  = (const int*)  d_in[1];
    const float* weight = (const float*)d_in[2];
    const float* wfact  = (const float*)d_in[3];
    const float* bias   = (const float*)d_in[4];
    const float* bfact  = (const float*)d_in[5];
    float*       out    = (float*)d_out;

    // workspace layout: [64 ints counts (padded)] [E*BTOK ints buckets]
    int* counts  = (int*)d_ws;
    int* buckets = counts + 64;

    hipMemsetAsync(counts, 0, 64 * sizeof(int), stream);
    bin_tokens_kernel<<<(BTOK + 255) / 256, 256, 0, stream>>>(route, counts, buckets);

    dim3 grid(E, OUT / CG);
    moe_wmma_kernel<<<grid, BDIM, LDS_BYTES, stream>>>(input, weight, wfact, bias,
                                                       bfact, counts, buckets, out);
}